// DisturbLabel_81638738362407
// MI455X (gfx1250) — compile-verified
//
#include <hip/hip_runtime.h>
#include <stdint.h>

#define TPB  256                 // 8 wave32s per workgroup
#define CCLS 32000               // number of classes (row length)
#define NVEC (CCLS / 4)          // 8000 float4 slots per row
#define FULL (NVEC / TPB)        // 31 uniform float4 trips per thread
#define TAIL (NVEC - FULL * TPB) // 64 leftover float4 slots

typedef int v4i __attribute__((ext_vector_type(4)));
typedef __attribute__((address_space(1))) v4i* gvec_p;   // global int4*
typedef __attribute__((address_space(3))) v4i* lvec_p;   // LDS int4*

// ---------------------------------------------------------------------------
// threefry2x32 block cipher (exact JAX implementation: 20 rounds, 5 key injections)
// ---------------------------------------------------------------------------
__device__ __forceinline__ uint32_t rotl32(uint32_t v, int n) {
  return (v << n) | (v >> (32 - n));
}

__device__ __forceinline__ void threefry2x32(uint32_t k0, uint32_t k1,
                                             uint32_t x0, uint32_t x1,
                                             uint32_t &o0, uint32_t &o1) {
  const uint32_t ks0 = k0, ks1 = k1, ks2 = 0x1BD11BDAu ^ k0 ^ k1;
  x0 += ks0; x1 += ks1;
#define TF_R(r) { x0 += x1; x1 = rotl32(x1, r); x1 ^= x0; }
  TF_R(13) TF_R(15) TF_R(26) TF_R(6)
  x0 += ks1; x1 += ks2 + 1u;
  TF_R(17) TF_R(29) TF_R(16) TF_R(24)
  x0 += ks2; x1 += ks0 + 2u;
  TF_R(13) TF_R(15) TF_R(26) TF_R(6)
  x0 += ks0; x1 += ks1 + 3u;
  TF_R(17) TF_R(29) TF_R(16) TF_R(24)
  x0 += ks1; x1 += ks2 + 4u;
  TF_R(13) TF_R(15) TF_R(26) TF_R(6)
  x0 += ks2; x1 += ks0 + 5u;
#undef TF_R
  o0 = x0; o1 = x1;
}

// Reproduce: key=jax.random.key(42); kr,kd=split(key);
// r=uniform(kr,(B,),f32); d=randint(kd,(B,),0,C-1);
// dlabel=d+(d>=tgt); new = (r < (C-1)/C*0.1) ? dlabel : tgt
__device__ __forceinline__ int disturbed_target(int b, int B, int tgt, int C) {
  // split(key(42)): counts iota(4) -> pairs (0,2),(1,3); kr=row0, kd=row1
  uint32_t kr0, kr1, kd0, kd1;
  threefry2x32(0u, 42u, 0u, 2u, kr0, kd0);
  threefry2x32(0u, 42u, 1u, 3u, kr1, kd1);

  // uniform bits: counts iota(B), halves pair (i, i+B/2)
  const uint32_t half = (uint32_t)(B >> 1);
  uint32_t o0, o1, rbits;
  if ((uint32_t)b < half) { threefry2x32(kr0, kr1, (uint32_t)b, (uint32_t)b + half, o0, o1); rbits = o0; }
  else                    { threefry2x32(kr0, kr1, (uint32_t)b - half, (uint32_t)b, o0, o1); rbits = o1; }
  const float r = __uint_as_float((rbits >> 9) | 0x3F800000u) - 1.0f;

  // randint bits: shape (2,B) -> counts iota(2B), pair (b, b+B): o0=higher, o1=lower
  uint32_t hi, lo;
  threefry2x32(kd0, kd1, (uint32_t)b, (uint32_t)(b + B), hi, lo);
  const uint32_t span = (uint32_t)(C - 1);
  uint32_t mult = 65536u % span;
  mult = (mult * mult) % span;                       // 2^32 mod span
  const uint32_t off = ((hi % span) * mult + (lo % span)) % span;
  const int d = (int)off;
  const int dlabel = d + ((d >= tgt) ? 1 : 0);

  const float bound = (float)(((double)C - 1.0) / (double)C * 0.1);
  return (r < bound) ? dlabel : tgt;
}

// ---------------------------------------------------------------------------
// Per-row NLL: stage full 125 KB row into LDS via async global->LDS copies
// (CDNA5 ASYNCcnt path), then exact two-pass softmax entirely out of LDS.
// HBM traffic = one read of the matrix (roofline minimum, ~45 us @ 23.3 TB/s).
// ---------------------------------------------------------------------------
__global__ __launch_bounds__(TPB) void disturb_nll_kernel(
    const float* __restrict__ logits, const int* __restrict__ target,
    float* __restrict__ row_nll, int B) {
  __shared__ float row[CCLS];          // 125 KB: fits CDNA5's 320 KB LDS/WGP
  __shared__ float wred[TPB / 32];

  const int b   = blockIdx.x;
  const int tid = threadIdx.x;
  const float* g = logits + (size_t)b * CCLS;

  // ---- Stage row into LDS: 31 uniform async B128 issues + 1 tail ----
#if defined(__gfx1250__) && __has_builtin(__builtin_amdgcn_global_load_async_to_lds_b128)
#pragma unroll
  for (int k = 0; k < FULL; ++k) {
    const int i = (tid + k * TPB) * 4;
    __builtin_amdgcn_global_load_async_to_lds_b128(
        (gvec_p)(g + i), (lvec_p)(&row[i]), 0, 0);
  }
  if (tid < TAIL) {
    const int i = (tid + FULL * TPB) * 4;
    __builtin_amdgcn_global_load_async_to_lds_b128(
        (gvec_p)(g + i), (lvec_p)(&row[i]), 0, 0);
  }
 #if __has_builtin(__builtin_amdgcn_s_wait_asynccnt)
  __builtin_amdgcn_s_wait_asynccnt(0);
 #else
  asm volatile("s_wait_asynccnt 0" ::: "memory");
 #endif
#else
  for (int i = tid; i < NVEC; i += TPB) {
    ((float4*)row)[i] = ((const float4*)g)[i];
  }
#endif
  __syncthreads();

  const float4* rowv = (const float4*)row;

  // ---- Pass 1: row max (float4 LDS reads -> ds_load_b128) ----
  float m = -3.402823466e+38f;
#pragma unroll 4
  for (int k = 0; k < FULL; ++k) {
    const float4 v = rowv[tid + k * TPB];
    m = fmaxf(m, fmaxf(fmaxf(v.x, v.y), fmaxf(v.z, v.w)));
  }
  if (tid < TAIL) {
    const float4 v = rowv[tid + FULL * TPB];
    m = fmaxf(m, fmaxf(fmaxf(v.x, v.y), fmaxf(v.z, v.w)));
  }
#pragma unroll
  for (int o = 16; o > 0; o >>= 1) m = fmaxf(m, __shfl_xor(m, o, 32));
  if ((tid & 31) == 0) wred[tid >> 5] = m;
  __syncthreads();
  float rowmax = wred[0];
#pragma unroll
  for (int w = 1; w < TPB / 32; ++w) rowmax = fmaxf(rowmax, wred[w]);
  __syncthreads();

  // ---- Pass 2: sum of exp(x - max) (float4 LDS reads) ----
  float s = 0.0f;
#pragma unroll 4
  for (int k = 0; k < FULL; ++k) {
    const float4 v = rowv[tid + k * TPB];
    s += __expf(v.x - rowmax) + __expf(v.y - rowmax) +
         __expf(v.z - rowmax) + __expf(v.w - rowmax);
  }
  if (tid < TAIL) {
    const float4 v = rowv[tid + FULL * TPB];
    s += __expf(v.x - rowmax) + __expf(v.y - rowmax) +
         __expf(v.z - rowmax) + __expf(v.w - rowmax);
  }
#pragma unroll
  for (int o = 16; o > 0; o >>= 1) s += __shfl_xor(s, o, 32);
  if ((tid & 31) == 0) wred[tid >> 5] = s;
  __syncthreads();

  if (tid == 0) {
    float tot = 0.0f;
#pragma unroll
    for (int w = 0; w < TPB / 32; ++w) tot += wred[w];
    const int nt = disturbed_target(b, B, target[b], CCLS);
    // nll = max + log(sum exp(x-max)) - x[new_target]
    row_nll[b] = rowmax + logf(tot) - row[nt];
  }
}

// ---------------------------------------------------------------------------
// Deterministic fixed-order mean reduction over B per-row NLLs (no atomics).
// ---------------------------------------------------------------------------
__global__ __launch_bounds__(TPB) void final_reduce_kernel(
    const float* __restrict__ row_nll, float* __restrict__ out, int B) {
  __shared__ float sdata[TPB];
  float s = 0.0f;
  for (int i = threadIdx.x; i < B; i += TPB) s += row_nll[i];
  sdata[threadIdx.x] = s;
  __syncthreads();
  for (int o = TPB / 2; o > 0; o >>= 1) {
    if (threadIdx.x < (unsigned)o) sdata[threadIdx.x] += sdata[threadIdx.x + o];
    __syncthreads();
  }
  if (threadIdx.x == 0) out[0] = sdata[0] / (float)B;
}

extern "C" void kernel_launch(void* const* d_in, const int* in_sizes, int n_in,
                              void* d_out, int out_size, void* d_ws, size_t ws_size,
                              hipStream_t stream) {
  const float* logits = (const float*)d_in[0];   // [B, 32000] f32
  const int*   target = (const int*)d_in[1];     // [B] int32 (JAX x64 off)
  const int B = in_sizes[1];

  float* row_nll = (float*)d_ws;                 // B floats of scratch

  disturb_nll_kernel<<<B, TPB, 0, stream>>>(logits, target, row_nll, B);
  final_reduce_kernel<<<1, TPB, 0, stream>>>(row_nll, (float*)d_out, B);
}